// SPIN_81071802680031
// MI455X (gfx1250) — compile-verified
//
#include <hip/hip_runtime.h>
#include <hip/hip_bf16.h>

typedef float v2f __attribute__((ext_vector_type(2)));
typedef float v4f __attribute__((ext_vector_type(4)));
typedef float v8f __attribute__((ext_vector_type(8)));

constexpr int B_ = 4, C_ = 64, H_ = 256, W_ = 256;
constexpr int SH = 16, SW = 16, NSPH = 16, NSPW = 16;
constexpr int S_ = NSPH * NSPW;   // 256 superpixels
constexpr int P_ = H_ * W_;       // 65536 pixels
constexpr float BIGF = 1e16f;

// ---------------------------------------------------------------------------
// K0: initial centroids = 16x16 block means. One WG (64 threads = 1 ch each)
// per (b, s). Grid channels are zero (M_PARAM=0) and dropped everywhere.
// ---------------------------------------------------------------------------
__global__ void k_block_mean(const float* __restrict__ x, float* __restrict__ spix) {
  int bs = blockIdx.x;                    // b*S + s
  int b = bs / S_, s = bs % S_;
  int r = s / NSPW, cb = s % NSPW;
  int c = threadIdx.x;                    // 64 threads = channels
  const float* xp = x + ((size_t)(b * C_ + c) * H_ + (size_t)r * SH) * W_ + (size_t)cb * SW;
  float acc = 0.f;
  for (int ly = 0; ly < SH; ++ly) {
    const v4f* row = (const v4f*)(xp + (size_t)ly * W_);
#pragma unroll
    for (int j = 0; j < SW / 4; ++j) { v4f v = row[j]; acc += v.x + v.y + v.z + v.w; }
  }
  spix[(size_t)bs * C_ + c] = acc * (1.0f / (SH * SW));
}

// ---------------------------------------------------------------------------
// Main kernel: one WG (256 thr = 8 waves) per (b, block). All 256 pixels of a
// block share the same 9 candidate stokens -> dist via WMMA f32 16x16x4:
//   dots[256x16] = X[256x64] * Scand^T[64x16]  (cands padded 9->16 with zeros)
// then per-pixel softmax over the 9 neighbors.
// MODE 0: fused sparse pooling -> atomicAdd partial num/den per candidate.
// MODE 1: store compact aff (B,9,P) for the dense streaming pass.
// ---------------------------------------------------------------------------
template <int MODE>
__global__ void k_aff(const float* __restrict__ x, const float* __restrict__ spix,
                      float* __restrict__ num, float* __restrict__ den,
                      float* __restrict__ affout) {
  extern __shared__ float smem[];
  float* xt   = smem;                 // [256][65] pixel tile, [p][c], pad 65
  float* sc   = xt + 256 * 65;        // [16][65]  candidate centroids
  float* dots = sc + 16 * 65;         // [256][17] dot products, reused for aff
  float* xsq  = dots + 256 * 17;      // [256] |x_p|^2
  float* ssq  = xsq + 256;            // [16]  |s_n|^2
  float* sval = ssq + 16;             // [16]  validity
  int*   scnd = (int*)(sval + 16);    // [16]  candidate s index

  int bs = blockIdx.x;
  int b = bs / S_, s = bs % S_;
  int r = s / NSPW, cb = s % NSPW;
  int t = threadIdx.x;
  int ly = t >> 4, lx = t & 15;
  int y = r * SH + ly, xx = cb * SW + lx;

  // load pixel tile (coalesced 64B rows per channel) + |x|^2
  {
    size_t base = ((size_t)b * C_ * H_ + (size_t)y) * W_ + xx;
    float a2 = 0.f;
    for (int c = 0; c < C_; ++c) {
      float v = x[base + (size_t)c * (H_ * W_)];
      xt[t * 65 + c] = v;
      a2 += v * v;
    }
    xsq[t] = a2;
  }
  // load 9 candidate centroids (rows 9..15 zero-padded)
  for (int i = t; i < 16 * C_; i += 256) {
    int n = i >> 6, c = i & 63;
    float v = 0.f;
    if (n < 9) {
      int dr = n / 3 - 1, dc = n % 3 - 1;
      int rr = r + dr, cc = cb + dc;
      if (rr >= 0 && rr < NSPH && cc >= 0 && cc < NSPW)
        v = spix[((size_t)b * S_ + (size_t)(rr * NSPW + cc)) * C_ + c];
    }
    sc[n * 65 + c] = v;
  }
  __syncthreads();
  if (t < 16) {
    float q = 0.f;
    for (int c = 0; c < C_; ++c) { float v = sc[t * 65 + c]; q += v * v; }
    ssq[t] = q;
    int ok = 0, cd = 0;
    if (t < 9) {
      int dr = t / 3 - 1, dc = t % 3 - 1;
      int rr = r + dr, cc = cb + dc;
      ok = (rr >= 0 && rr < NSPH && cc >= 0 && cc < NSPW);
      cd = rr * NSPW + cc;
    }
    sval[t] = ok ? 1.f : 0.f;
    scnd[t] = cd;
  }
  __syncthreads();

  // WMMA stage: 8 waves x 2 M-tiles = 16 tiles of 16 pixels
  {
    int w = t >> 5, lane = t & 31;
    int ml = lane & 15, hi = lane >> 4;     // A: lanes 0-15 K={0,1}, 16-31 K={2,3}
    for (int tt = 0; tt < 2; ++tt) {
      int tile = w * 2 + tt;
      int p0 = tile * 16 + ml;
      v8f acc = {0.f, 0.f, 0.f, 0.f, 0.f, 0.f, 0.f, 0.f};
#pragma unroll
      for (int k0 = 0; k0 < C_; k0 += 4) {
        int ca = k0 + 2 * hi;
        v2f av, bv;
        av.x = xt[p0 * 65 + ca];  av.y = xt[p0 * 65 + ca + 1];
        bv.x = sc[ml * 65 + ca];  bv.y = sc[ml * 65 + ca + 1];
        acc = __builtin_amdgcn_wmma_f32_16x16x4_f32(
            false, av, false, bv, (short)0, acc, false, false);
      }
#pragma unroll
      for (int v = 0; v < 8; ++v)          // C layout: M = v + 8*hi, N = ml
        dots[(tile * 16 + v + 8 * hi) * 17 + ml] = acc[v];
    }
  }
  __syncthreads();

  // softmax over 9 neighbors (invalid -> exactly 0, matching f32 underflow)
  float aff[9];
  {
    float xs = xsq[t];
    float dneg[9];
    float mx = -3.4e38f;
#pragma unroll
    for (int n = 0; n < 9; ++n) {
      float dd = (sval[n] > 0.f) ? (xs + ssq[n] - 2.f * dots[t * 17 + n]) : BIGF;
      dneg[n] = -dd;
      mx = fmaxf(mx, dneg[n]);
    }
    float ssum = 0.f;
#pragma unroll
    for (int n = 0; n < 9; ++n) {
      float e = (sval[n] > 0.f) ? __expf(dneg[n] - mx) : 0.f;
      aff[n] = e; ssum += e;
    }
    float inv = 1.f / ssum;
#pragma unroll
    for (int n = 0; n < 9; ++n) aff[n] *= inv;
  }

  if (MODE == 0) {
    // fused sparse pooling: per-candidate partial sums over this block's pixels
    __syncthreads();                       // all dots reads done
#pragma unroll
    for (int n = 0; n < 9; ++n) dots[t * 17 + n] = aff[n];
    __syncthreads();
    for (int i = t; i < 9 * 65; i += 256) {
      int n = i / 65, j = i % 65;
      if (sval[n] == 0.f) continue;
      float acc = 0.f;
      if (j < C_) {
        for (int p = 0; p < 256; ++p) acc += dots[p * 17 + n] * xt[p * 65 + j];
        atomicAdd(&num[((size_t)b * S_ + scnd[n]) * C_ + j], acc);
      } else {
        for (int p = 0; p < 256; ++p) acc += dots[p * 17 + n];
        atomicAdd(&den[b * S_ + scnd[n]], acc);
      }
    }
  } else {
    size_t p = (size_t)y * W_ + xx;
#pragma unroll
    for (int n = 0; n < 9; ++n)
      affout[((size_t)(b * 9 + n) << 16) + p] = aff[n];
  }
}

// ---------------------------------------------------------------------------
// K3: centroid update  spix[b][s][c] = num / (den + 1e-16)
// ---------------------------------------------------------------------------
__global__ void k_update(const float* __restrict__ num, const float* __restrict__ den,
                         float* __restrict__ spix) {
  int i = blockIdx.x * blockDim.x + threadIdx.x;
  if (i >= B_ * S_ * C_) return;
  spix[i] = num[i] / (den[i >> 6] + 1e-16f);
}

// ---------------------------------------------------------------------------
// K5: densify to (B,S,P): for each row (b,s), nonzeros only where pixel's
// block is in the 3x3 neighborhood of s. Non-temporal float4 streaming.
// ---------------------------------------------------------------------------
__global__ void k_out(const float* __restrict__ aff, float* __restrict__ out) {
  int wg = blockIdx.x;                    // (b*S+s)*4 + seg
  int seg = wg & 3;
  int bs = wg >> 2;
  int b = bs / S_, s = bs % S_;
  int r = s / NSPW, cb = s % NSPW;
  int t = threadIdx.x;
  float* orow = out + ((size_t)bs << 16);
  const float* abase = aff + (((size_t)b * 9) << 16);
  for (int j = 0; j < 16; ++j) {
    int q = (seg << 14) + ((j * 256 + t) << 2);   // 4 pixels, same 16x16 block
    int yy = q >> 8, xq = q & 255;
    int dr = r - (yy >> 4), dc = cb - (xq >> 4);
    v4f v = {0.f, 0.f, 0.f, 0.f};
    if (dr >= -1 && dr <= 1 && dc >= -1 && dc <= 1) {
      int k = (dr + 1) * 3 + (dc + 1);
      v = *(const v4f*)(abase + ((size_t)k << 16) + q);
    }
    __builtin_nontemporal_store(v, (v4f*)(orow + q));
  }
}

// ---------------------------------------------------------------------------
extern "C" void kernel_launch(void* const* d_in, const int* in_sizes, int n_in,
                              void* d_out, int out_size, void* d_ws, size_t ws_size,
                              hipStream_t stream) {
  (void)in_sizes; (void)n_in; (void)out_size; (void)ws_size;
  const float* x = (const float*)d_in[0];   // (4,64,256,256) f32; stoken sizes fixed 16

  float* ws   = (float*)d_ws;
  float* spix = ws;                          // B*S*C   = 65536
  float* num  = spix + B_ * S_ * C_;         // B*S*C   = 65536
  float* den  = num + B_ * S_ * C_;          // B*S     = 1024
  float* affb = den + B_ * S_;               // B*9*P   = 2359296
  float* out  = (float*)d_out;

  size_t lds = (size_t)(256 * 65 + 16 * 65 + 256 * 17 + 256 + 16 + 16 + 16) * 4; // ~89 KB

  hipMemsetAsync(num, 0, (size_t)(B_ * S_ * C_ + B_ * S_) * sizeof(float), stream);
  k_block_mean<<<B_ * S_, C_, 0, stream>>>(x, spix);
  k_aff<0><<<B_ * S_, 256, lds, stream>>>(x, spix, num, den, nullptr);  // iter 0 + pooling
  k_update<<<(B_ * S_ * C_ + 255) / 256, 256, 0, stream>>>(num, den, spix);
  k_aff<1><<<B_ * S_, 256, lds, stream>>>(x, spix, num, den, affb);     // iter 1 -> compact aff
  k_out<<<B_ * S_ * 4, 256, 0, stream>>>(affb, out);                    // 256 MB NT stream
}